// Sparsemax_206158430852
// MI455X (gfx1250) — compile-verified
//
#include <hip/hip_runtime.h>
#include <hip/hip_bf16.h>

// Sparsemax along last axis. x: (8, 4096, 1024) fp32 -> 32768 rows of 1024.
// One wave32 per row; 32 elements per lane held in VGPRs.
// Threshold tau found with Michelot's fixed-point iteration (exact, no sort).
// Streaming kernel: non-temporal b128 loads/stores + global_prefetch_b8 of the
// next block's row (lane l covers cacheline l*128 -> 32 lanes = full 4KB row).

#define SPX_N          1024
#define SPX_EPL        32          // elements per lane (N / 32)
#define SPX_V4PL       8           // float4 loads per lane
#define ROWS_PER_BLOCK 8           // 8 waves of 32 per block

typedef float v4f __attribute__((ext_vector_type(4)));

__global__ __launch_bounds__(256)
void Sparsemax_206158430852_kernel(const float* __restrict__ x,
                                   float* __restrict__ out,
                                   int nrows) {
    const int lane = threadIdx.x & 31;
    const int wave = threadIdx.x >> 5;
    const int row  = blockIdx.x * ROWS_PER_BLOCK + wave;
    if (row >= nrows) return;

    const v4f* __restrict__ xr4 = reinterpret_cast<const v4f*>(x + (size_t)row * SPX_N);
    v4f* __restrict__       or4 = reinterpret_cast<v4f*>(out + (size_t)row * SPX_N);

    // gfx1250 prefetch path (global_prefetch_b8): pull the next block's row
    // toward L2 while we crunch this one.
    {
        const int nrow = row + ROWS_PER_BLOCK;
        if (nrow < nrows) {
            const char* p = reinterpret_cast<const char*>(x) +
                            (size_t)nrow * SPX_N * sizeof(float) + (size_t)lane * 128;
            __builtin_prefetch(p, 0, 1);
        }
    }

    // ---- load 32 elements per lane (8 x b128 NT, coalesced 512B per instr) ----
    float z[SPX_EPL];
#pragma unroll
    for (int j = 0; j < SPX_V4PL; ++j) {
        v4f t = __builtin_nontemporal_load(&xr4[lane + 32 * j]);
        z[4 * j + 0] = t.x;
        z[4 * j + 1] = t.y;
        z[4 * j + 2] = t.z;
        z[4 * j + 3] = t.w;
    }

    // ---- row max (lane-local then wave shuffle tree) ----
    float m = z[0];
#pragma unroll
    for (int i = 1; i < SPX_EPL; ++i) m = fmaxf(m, z[i]);
#pragma unroll
    for (int off = 16; off >= 1; off >>= 1) m = fmaxf(m, __shfl_xor(m, off, 32));

    // ---- z = x - max, total sum ----
    float s = 0.0f;
#pragma unroll
    for (int i = 0; i < SPX_EPL; ++i) {
        z[i] -= m;
        s += z[i];
    }
#pragma unroll
    for (int off = 16; off >= 1; off >>= 1) s += __shfl_xor(s, off, 32);

    // ---- Michelot fixed-point for tau ----
    // tau_0 over full set; each iteration shrinks the active set {z > tau}.
    // tau is monotone nondecreasing; equal counts => identical sets => fixed point.
    float tau  = (s - 1.0f) * (1.0f / (float)SPX_N);
    int   cprv = SPX_N;
    for (int it = 0; it <= SPX_N; ++it) {
        float ls  = 0.0f;
        float lcf = 0.0f;
#pragma unroll
        for (int i = 0; i < SPX_EPL; ++i) {
            const bool a = z[i] > tau;
            ls  += a ? z[i] : 0.0f;
            lcf += a ? 1.0f : 0.0f;
        }
#pragma unroll
        for (int off = 16; off >= 1; off >>= 1) {
            ls  += __shfl_xor(ls, off, 32);
            lcf += __shfl_xor(lcf, off, 32);
        }
        const int c = (int)lcf;            // exact: c <= 1024 fits fp32
        if (c == cprv) break;              // wave-uniform (post-reduction)
        tau  = (ls - 1.0f) / lcf;
        cprv = c;
    }

    // ---- output: max(z - tau, 0), 8 x b128 NT stores ----
#pragma unroll
    for (int j = 0; j < SPX_V4PL; ++j) {
        v4f o;
        o.x = fmaxf(z[4 * j + 0] - tau, 0.0f);
        o.y = fmaxf(z[4 * j + 1] - tau, 0.0f);
        o.z = fmaxf(z[4 * j + 2] - tau, 0.0f);
        o.w = fmaxf(z[4 * j + 3] - tau, 0.0f);
        __builtin_nontemporal_store(o, &or4[lane + 32 * j]);
    }
}

extern "C" void kernel_launch(void* const* d_in, const int* in_sizes, int n_in,
                              void* d_out, int out_size, void* d_ws, size_t ws_size,
                              hipStream_t stream) {
    const float* x   = (const float*)d_in[0];
    float*       out = (float*)d_out;
    const int nrows  = in_sizes[0] / SPX_N;            // 8*4096 = 32768
    const int blocks = (nrows + ROWS_PER_BLOCK - 1) / ROWS_PER_BLOCK;
    Sparsemax_206158430852_kernel<<<blocks, 256, 0, stream>>>(x, out, nrows);
}